// DLGCNLayer_43164421325127
// MI455X (gfx1250) — compile-verified
//
#include <hip/hip_runtime.h>

// ---------------------------------------------------------------------------
// DLGCN layer for MI455X (gfx1250, wave32).
//   Phase 1: 4x COO SpMM via gather + global f32 atomic scatter (L2-resident).
//   Phase 2: 2x dense GEMM [rows,256]x[256,128] via V_WMMA_F32_16X16X4_F32
//            (exact fp32 matrix pipe) + bias + ReLU.
// ---------------------------------------------------------------------------

typedef __attribute__((ext_vector_type(2))) float v2f;
typedef __attribute__((ext_vector_type(8))) float v8f;

#define N_USERS 100000
#define N_ITEMS 50000
#define DFEAT   128
#define NEDGE   2000000

// ---------------------------------------------------------------------------
// Zero-fill workspace accumulators (grid-stride).
// ---------------------------------------------------------------------------
__global__ void zero_kernel(float* __restrict__ p, long n) {
    long i = (long)blockIdx.x * blockDim.x + threadIdx.x;
    long stride = (long)gridDim.x * blockDim.x;
    for (; i < n; i += stride) p[i] = 0.0f;
}

// ---------------------------------------------------------------------------
// SpMM scatter: out[rows[e], :] += vals[e] * X[cols[e], :]
// One wave32 per edge; each lane owns 4 contiguous floats (float4 gather,
// 4x global_atomic_add_f32 scatter). 8 edges per 256-thread block.
// ---------------------------------------------------------------------------
__global__ void __launch_bounds__(256)
spmm_scatter(const int* __restrict__ rows, const int* __restrict__ cols,
             const float* __restrict__ vals, const float* __restrict__ X,
             float* __restrict__ out, int nEdges) {
    int lane = threadIdx.x & 31;
    int edge = blockIdx.x * (blockDim.x >> 5) + (threadIdx.x >> 5);
    if (edge >= nEdges) return;

    int r = rows[edge];
    int c = cols[edge];
    float v = vals[edge];

    const float4* xrow = (const float4*)(X + (size_t)c * DFEAT);
    float4 x = xrow[lane];                       // 16B coalesced gather
    float* o = out + (size_t)r * DFEAT + lane * 4;
    atomicAdd(o + 0, v * x.x);
    atomicAdd(o + 1, v * x.y);
    atomicAdd(o + 2, v * x.z);
    atomicAdd(o + 3, v * x.w);
}

// ---------------------------------------------------------------------------
// GEMM + bias + ReLU:
//   Out[r, n] = relu( sum_k concat(Ho, Fea)[r, k] * W[n, k] + b[n] )
// Block = 256 threads = 8 waves; block tile = 16 rows x 128 cols;
// wave w computes the 16x16 n-tile at nBase = 16*w. K = 256, stepped by 4
// with V_WMMA_F32_16X16X4_F32. A tile staged in LDS (stride 260 floats ->
// conflict-free b64 reads across all 64 banks). nRows % 16 == 0 guaranteed,
// so EXEC is all-ones at every WMMA.
// ---------------------------------------------------------------------------
#define A_STRIDE 260   // 16 rows * 260 floats = 16.25 KB LDS

__global__ void __launch_bounds__(256)
gemm_bias_relu(const float* __restrict__ Ho, const float* __restrict__ Fea,
               const float* __restrict__ W, const float* __restrict__ bias,
               float* __restrict__ Out) {
    __shared__ float Atile[16 * A_STRIDE];

    const int rowBase = blockIdx.x * 16;
    const int tid = threadIdx.x;

    // Cooperative fill: 16 rows x 256 cols, float4-vectorized, building the
    // [Ho | Fea] concatenation on the fly. 1024 float4 slots / 256 threads.
    for (int idx = tid; idx < 16 * 64; idx += 256) {
        int r  = idx >> 6;          // 0..15
        int c4 = idx & 63;          // 0..63
        int c  = c4 * 4;            // 0..252
        size_t gr = (size_t)(rowBase + r);
        const float* src = (c < DFEAT) ? (Ho + gr * DFEAT + c)
                                       : (Fea + gr * DFEAT + (c - DFEAT));
        float4 val = *(const float4*)src;
        *(float4*)&Atile[r * A_STRIDE + c] = val;
    }
    __syncthreads();

    const int wave  = tid >> 5;          // 0..7 -> n-tile
    const int lane  = tid & 31;
    const int nBase = wave * 16;
    const int m     = lane & 15;         // A row (M) == B col (N) lane index
    const int khalf = (lane >> 4) * 2;   // lanes 16-31 hold K+2, K+3

    // B fragment source: B[k][n] = W[n*256 + k]  (W is [128, 256] row-major)
    const float* wrow = W + (size_t)(nBase + m) * 256;
    const float* arow = &Atile[m * A_STRIDE];

    v8f acc = {};
    #pragma unroll 8
    for (int k0 = 0; k0 < 256; k0 += 4) {
        int kk = k0 + khalf;
        v2f a = *(const v2f*)(arow + kk);   // A 16x4 fragment (ISA layout)
        v2f b = *(const v2f*)(wrow + kk);   // B 4x16 fragment
        acc = __builtin_amdgcn_wmma_f32_16x16x4_f32(
            /*neg_a=*/false, a, /*neg_b=*/false, b,
            /*c_mod=*/(short)0, acc, /*reuse_a=*/false, /*reuse_b=*/false);
    }

    // C/D layout: VGPR v -> rows (v, v+8); N = lane & 15.
    const float bv = bias[nBase + m];
    const int rowOff = (lane >> 4) * 8;
    #pragma unroll
    for (int v = 0; v < 8; ++v) {
        int r = rowBase + v + rowOff;
        float val = acc[v] + bv;
        Out[(size_t)r * DFEAT + nBase + m] = fmaxf(val, 0.0f);
    }
}

// ---------------------------------------------------------------------------
// Launch. Inputs (setup_inputs order):
//   0 ufea [100000,128] f32      1 vfea [50000,128] f32
//   2 edge_u [2M] i32            3 edge_v [2M] i32       4 edge_val [2M] f32
//   5 user_W [128,256] f32       6 user_b [128] f32
//   7 item_W [128,256] f32       8 item_b [128] f32
// Output: User [100000,128] then Item [50000,128], f32.
// Workspace (floats): tmpItems 6.4M | tmpUsers 12.8M | UserHo 12.8M | ItemHo 6.4M
// ---------------------------------------------------------------------------
extern "C" void kernel_launch(void* const* d_in, const int* in_sizes, int n_in,
                              void* d_out, int out_size, void* d_ws, size_t ws_size,
                              hipStream_t stream) {
    const float* ufea     = (const float*)d_in[0];
    const float* vfea     = (const float*)d_in[1];
    const int*   edge_u   = (const int*)d_in[2];
    const int*   edge_v   = (const int*)d_in[3];
    const float* edge_val = (const float*)d_in[4];
    const float* user_W   = (const float*)d_in[5];
    const float* user_b   = (const float*)d_in[6];
    const float* item_W   = (const float*)d_in[7];
    const float* item_b   = (const float*)d_in[8];

    float* ws = (float*)d_ws;
    float* tmpItems = ws;                                   // [N_ITEMS, 128]
    float* tmpUsers = tmpItems + (size_t)N_ITEMS * DFEAT;   // [N_USERS, 128]
    float* UserHo   = tmpUsers + (size_t)N_USERS * DFEAT;   // [N_USERS, 128]
    float* ItemHo   = UserHo   + (size_t)N_USERS * DFEAT;   // [N_ITEMS, 128]
    const long accTotal = (long)(2 * N_ITEMS + 2 * N_USERS) * DFEAT;

    float* outUser = (float*)d_out;
    float* outItem = outUser + (size_t)N_USERS * DFEAT;

    // 0) zero all scatter accumulators (fresh every call: deterministic)
    zero_kernel<<<2048, 256, 0, stream>>>(ws, accTotal);

    const int edgeBlocks = (NEDGE + 7) / 8;   // 8 edges (waves) per block

    // 1) S1: tmpItems = VU_adj @ ufea   (rows=edge_v, cols=edge_u)
    spmm_scatter<<<edgeBlocks, 256, 0, stream>>>(edge_v, edge_u, edge_val,
                                                 ufea, tmpItems, NEDGE);
    // 2) S3: tmpUsers = UV_adj @ vfea   (rows=edge_u, cols=edge_v)
    spmm_scatter<<<edgeBlocks, 256, 0, stream>>>(edge_u, edge_v, edge_val,
                                                 vfea, tmpUsers, NEDGE);
    // 3) S2: UserHo = UV_adj @ tmpItems (depends on S1)
    spmm_scatter<<<edgeBlocks, 256, 0, stream>>>(edge_u, edge_v, edge_val,
                                                 tmpItems, UserHo, NEDGE);
    // 4) S4: ItemHo = VU_adj @ tmpUsers (depends on S3)
    spmm_scatter<<<edgeBlocks, 256, 0, stream>>>(edge_v, edge_u, edge_val,
                                                 tmpUsers, ItemHo, NEDGE);

    // 5) User = relu(concat(UserHo, ufea) @ user_W^T + user_b)
    gemm_bias_relu<<<N_USERS / 16, 256, 0, stream>>>(UserHo, ufea, user_W,
                                                     user_b, outUser);
    // 6) Item = relu(concat(ItemHo, vfea) @ item_W^T + item_b)
    gemm_bias_relu<<<N_ITEMS / 16, 256, 0, stream>>>(ItemHo, vfea, item_W,
                                                     item_b, outItem);
}